// GNN_node_30279519437414
// MI455X (gfx1250) — compile-verified
//
#include <hip/hip_runtime.h>

#define N_INST 200000
#define N_NET  50000
#define NTOT   (N_INST + N_NET)
#define NEDGE  2000000
#define LAYERS 3

typedef float v2f __attribute__((ext_vector_type(2)));
typedef float v8f __attribute__((ext_vector_type(8)));

__device__ __forceinline__ float leaky_f(float x) { return x > 0.f ? x : 0.1f * x; }

// ---------------- encoders (small GEMMs, VALU; traffic-trivial) ----------------

// t[node,128] = leaky(x[node,16] @ W1[16,128] + b1)
__global__ void enc1_inst(const float* __restrict__ x, const float* __restrict__ W1,
                          const float* __restrict__ b1, float* __restrict__ t) {
  int gid = blockIdx.x * blockDim.x + threadIdx.x;
  int node = gid >> 5;
  if (node >= N_INST) return;
  int j0 = (gid & 31) << 2;
  const float* xr = x + (size_t)node * 16;
  float xv[16];
#pragma unroll
  for (int i = 0; i < 16; ++i) xv[i] = xr[i];
  float a0 = b1[j0], a1 = b1[j0 + 1], a2 = b1[j0 + 2], a3 = b1[j0 + 3];
#pragma unroll
  for (int i = 0; i < 16; ++i) {
    float4 w = *(const float4*)(W1 + (size_t)i * 128 + j0);
    a0 += xv[i] * w.x; a1 += xv[i] * w.y; a2 += xv[i] * w.z; a3 += xv[i] * w.w;
  }
  float* tp = t + (size_t)node * 128 + j0;
  tp[0] = leaky_f(a0); tp[1] = leaky_f(a1); tp[2] = leaky_f(a2); tp[3] = leaky_f(a3);
}

// h0 slice of out: out[node,0:64] = leaky(t @ W2[128,64] + b2)   (out row stride 256)
__global__ void enc2_inst(const float* __restrict__ t, const float* __restrict__ W2,
                          const float* __restrict__ b2, float* __restrict__ out) {
  int gid = blockIdx.x * blockDim.x + threadIdx.x;
  int node = gid >> 4;
  if (node >= N_INST) return;
  int d0 = (gid & 15) << 2;
  float a0 = b2[d0], a1 = b2[d0 + 1], a2 = b2[d0 + 2], a3 = b2[d0 + 3];
  const float* tp = t + (size_t)node * 128;
#pragma unroll 4
  for (int j = 0; j < 128; j += 4) {
    float4 tv = *(const float4*)(tp + j);
    float4 w0 = *(const float4*)(W2 + (size_t)(j + 0) * 64 + d0);
    float4 w1 = *(const float4*)(W2 + (size_t)(j + 1) * 64 + d0);
    float4 w2 = *(const float4*)(W2 + (size_t)(j + 2) * 64 + d0);
    float4 w3 = *(const float4*)(W2 + (size_t)(j + 3) * 64 + d0);
    a0 += tv.x * w0.x + tv.y * w1.x + tv.z * w2.x + tv.w * w3.x;
    a1 += tv.x * w0.y + tv.y * w1.y + tv.z * w2.y + tv.w * w3.y;
    a2 += tv.x * w0.z + tv.y * w1.z + tv.z * w2.z + tv.w * w3.z;
    a3 += tv.x * w0.w + tv.y * w1.w + tv.z * w2.w + tv.w * w3.w;
  }
  float4 r; r.x = leaky_f(a0); r.y = leaky_f(a1); r.z = leaky_f(a2); r.w = leaky_f(a3);
  *(float4*)(out + (size_t)node * 256 + d0) = r;
}

// tnet[node,64] = leaky(x_net[node,8] @ W1[8,64] + b1)
__global__ void enc1_net(const float* __restrict__ xn, const float* __restrict__ W1,
                         const float* __restrict__ b1, float* __restrict__ t) {
  int gid = blockIdx.x * blockDim.x + threadIdx.x;
  int node = gid >> 4;
  if (node >= N_NET) return;
  int j0 = (gid & 15) << 2;
  const float* xr = xn + (size_t)node * 8;
  float a0 = b1[j0], a1 = b1[j0 + 1], a2 = b1[j0 + 2], a3 = b1[j0 + 3];
#pragma unroll
  for (int i = 0; i < 8; ++i) {
    float xv = xr[i];
    float4 w = *(const float4*)(W1 + (size_t)i * 64 + j0);
    a0 += xv * w.x; a1 += xv * w.y; a2 += xv * w.z; a3 += xv * w.w;
  }
  float* tp = t + (size_t)node * 64 + j0;
  tp[0] = leaky_f(a0); tp[1] = leaky_f(a1); tp[2] = leaky_f(a2); tp[3] = leaky_f(a3);
}

// h0 rows N_INST.. : out[(N_INST+node),0:64] = leaky(tnet @ W2[64,64] + b2)
__global__ void enc2_net(const float* __restrict__ t, const float* __restrict__ W2,
                         const float* __restrict__ b2, float* __restrict__ out) {
  int gid = blockIdx.x * blockDim.x + threadIdx.x;
  int node = gid >> 4;
  if (node >= N_NET) return;
  int d0 = (gid & 15) << 2;
  float a0 = b2[d0], a1 = b2[d0 + 1], a2 = b2[d0 + 2], a3 = b2[d0 + 3];
  const float* tp = t + (size_t)node * 64;
#pragma unroll 4
  for (int j = 0; j < 64; j += 4) {
    float4 tv = *(const float4*)(tp + j);
    float4 w0 = *(const float4*)(W2 + (size_t)(j + 0) * 64 + d0);
    float4 w1 = *(const float4*)(W2 + (size_t)(j + 1) * 64 + d0);
    float4 w2 = *(const float4*)(W2 + (size_t)(j + 2) * 64 + d0);
    float4 w3 = *(const float4*)(W2 + (size_t)(j + 3) * 64 + d0);
    a0 += tv.x * w0.x + tv.y * w1.x + tv.z * w2.x + tv.w * w3.x;
    a1 += tv.x * w0.y + tv.y * w1.y + tv.z * w2.y + tv.w * w3.y;
    a2 += tv.x * w0.z + tv.y * w1.z + tv.z * w2.z + tv.w * w3.z;
    a3 += tv.x * w0.w + tv.y * w1.w + tv.z * w2.w + tv.w * w3.w;
  }
  float4 r; r.x = leaky_f(a0); r.y = leaky_f(a1); r.z = leaky_f(a2); r.w = leaky_f(a3);
  *(float4*)(out + (size_t)(N_INST + node) * 256 + d0) = r;
}

// ---------------- degrees ----------------
__global__ void deg_init(float* __restrict__ degf, float* __restrict__ degr) {
  int n = blockIdx.x * blockDim.x + threadIdx.x;
  if (n < NTOT) { degf[n] = 1.f; degr[n] = 1.f; }
}
__global__ void deg_accum(const int* __restrict__ row, const int* __restrict__ col,
                          float* __restrict__ degf, float* __restrict__ degr) {
  int e = blockIdx.x * blockDim.x + threadIdx.x;
  if (e < NEDGE) { atomicAdd(degf + row[e], 1.f); atomicAdd(degr + col[e], 1.f); }
}
__global__ void deg_fin(const float* __restrict__ degf, const float* __restrict__ degr,
                        float* __restrict__ disf, float* __restrict__ disr,
                        float* __restrict__ invf, float* __restrict__ invr) {
  int n = blockIdx.x * blockDim.x + threadIdx.x;
  if (n < NTOT) {
    float a = degf[n]; disf[n] = rsqrtf(a); invf[n] = 1.f / a;
    float c = degr[n]; disr[n] = rsqrtf(c); invr[n] = 1.f / c;
  }
}

// ---------------- GCN linear via fp32 WMMA: x = h@W + b ; agg (=|+=) relu(x+root)/deg ----
// One wave -> 16x64 output tile. hbase has row stride hstride (=256: a d_out slice).
__global__ __launch_bounds__(128) void conv_gemm(
    const float* __restrict__ hbase, int hstride,
    const float* __restrict__ W, const float* __restrict__ bias,
    const float* __restrict__ root, const float* __restrict__ invdeg,
    float* __restrict__ xout, float* __restrict__ agg, int addmode) {
  __shared__ float sW[64 * 65];  // row stride 65 avoids half-wave bank conflicts
  __shared__ float sB[64];
  __shared__ float sR[64];
  int tid = threadIdx.x;
  for (int i = tid; i < 64 * 64; i += 128) sW[(i >> 6) * 65 + (i & 63)] = W[i];
  if (tid < 64) { sB[tid] = bias[tid]; sR[tid] = root[tid]; }
  __syncthreads();

  int wave = tid >> 5, lane = tid & 31;
  int m0 = (blockIdx.x * 4 + wave) * 16;
  if (m0 >= NTOT) return;

  int arow = lane & 15;          // M within tile for A frag
  int koff = (lane >> 4) << 1;   // 0 or 2: which K pair this half-wave holds
  int col  = lane & 15;          // N within 16-wide tile for B/C frags
  const float* hrow = hbase + (size_t)(m0 + arow) * (size_t)hstride;

  v8f acc[4] = {v8f{}, v8f{}, v8f{}, v8f{}};
#pragma unroll 4
  for (int k = 0; k < 64; k += 4) {
    v2f a;
    a.x = hrow[k + koff];
    a.y = hrow[k + koff + 1];
    const float* w0 = sW + (size_t)(k + koff) * 65 + col;
    const float* w1 = w0 + 65;
#pragma unroll
    for (int nt = 0; nt < 4; ++nt) {
      v2f b;
      b.x = w0[nt * 16];
      b.y = w1[nt * 16];
      acc[nt] = __builtin_amdgcn_wmma_f32_16x16x4_f32(
          false, a, false, b, (short)0, acc[nt], false, false);
    }
  }

  // C/D layout: VGPR r -> lanes 0-15: M=r, lanes 16-31: M=r+8; N = lane&15
  int mb = m0 + ((lane >> 4) << 3);
#pragma unroll
  for (int nt = 0; nt < 4; ++nt) {
    int c = nt * 16 + col;
    float bb = sB[c], rr = sR[c];
#pragma unroll
    for (int r = 0; r < 8; ++r) {
      int m = mb + r;
      float xv = acc[nt][r] + bb;
      size_t off = (size_t)m * 64 + c;
      xout[off] = xv;
      float s = fmaxf(xv + rr, 0.f) * invdeg[m];
      if (addmode) agg[off] += s; else agg[off] = s;
    }
  }
}

// ---------------- edge phase: agg[dst] += dis[src]*dis[dst] * relu(x[src]) -----------
__global__ void edge_scatter(const float* __restrict__ x, float* __restrict__ agg,
                             const int* __restrict__ src, const int* __restrict__ dst,
                             const float* __restrict__ dis) {
  int gid = blockIdx.x * blockDim.x + threadIdx.x;
  int e = gid >> 4;
  if (e >= NEDGE) return;
  int part = (gid & 15) << 2;
  int s = src[e], d = dst[e];
  float nrm = dis[s] * dis[d];
  float4 xv = *(const float4*)(x + (size_t)s * 64 + part);
  float* ap = agg + (size_t)d * 64 + part;
  atomicAdd(ap + 0, nrm * fmaxf(xv.x, 0.f));
  atomicAdd(ap + 1, nrm * fmaxf(xv.y, 0.f));
  atomicAdd(ap + 2, nrm * fmaxf(xv.z, 0.f));
  atomicAdd(ap + 3, nrm * fmaxf(xv.w, 0.f));
}

// ---------------- LayerNorm + leaky, write slice of out --------------------------------
__global__ void finalize(const float* __restrict__ agg, const float* __restrict__ g,
                         const float* __restrict__ bta, float* __restrict__ out, int slice) {
  int gw = (blockIdx.x * blockDim.x + threadIdx.x) >> 5;
  int lane = threadIdx.x & 31;
  if (gw >= NTOT) return;
  float2 v = *(const float2*)(agg + (size_t)gw * 64 + (lane << 1));
  float s = v.x + v.y;
#pragma unroll
  for (int o = 16; o; o >>= 1) s += __shfl_xor(s, o, 32);
  float mean = s * (1.f / 64.f);
  float dx = v.x - mean, dy = v.y - mean;
  float q = dx * dx + dy * dy;
#pragma unroll
  for (int o = 16; o; o >>= 1) q += __shfl_xor(q, o, 32);
  float rstd = rsqrtf(q * (1.f / 64.f) + 1e-5f);
  int d0 = lane << 1;
  float2 r;
  r.x = leaky_f(dx * rstd * g[d0] + bta[d0]);
  r.y = leaky_f(dy * rstd * g[d0 + 1] + bta[d0 + 1]);
  *(float2*)(out + (size_t)gw * 256 + slice * 64 + d0) = r;
}

// ---------------- driver ----------------
extern "C" void kernel_launch(void* const* d_in, const int* in_sizes, int n_in,
                              void* d_out, int out_size, void* d_ws, size_t ws_size,
                              hipStream_t stream) {
  const float* x       = (const float*)d_in[0];
  const float* x_net   = (const float*)d_in[1];
  const int*   eidx    = (const int*)d_in[2];
  const float* enc1_W  = (const float*)d_in[3];
  const float* enc1_b  = (const float*)d_in[4];
  const float* enc2_W  = (const float*)d_in[5];
  const float* enc2_b  = (const float*)d_in[6];
  const float* net1_W  = (const float*)d_in[7];
  const float* net1_b  = (const float*)d_in[8];
  const float* net2_W  = (const float*)d_in[9];
  const float* net2_b  = (const float*)d_in[10];
  const float* conv_W  = (const float*)d_in[11];
  const float* conv_b  = (const float*)d_in[12];
  const float* conv_rt = (const float*)d_in[13];
  const float* re_W    = (const float*)d_in[14];
  const float* re_b    = (const float*)d_in[15];
  const float* re_rt   = (const float*)d_in[16];
  const float* ln_g    = (const float*)d_in[17];
  const float* ln_b    = (const float*)d_in[18];
  float* out = (float*)d_out;
  char*  ws  = (char*)d_ws;

  // workspace layout (encoder temps alias the conv-phase buffers; disjoint in time)
  float* xbuf = (float*)(ws);                      // NTOT*64 f32 = 64 MB
  float* agg  = (float*)(ws + 64000000ULL);        // NTOT*64 f32 = 64 MB
  float* tbuf = (float*)(ws);                      // N_INST*128 f32 = 102.4 MB (alias)
  float* tnet = (float*)(ws + 102400000ULL);       // N_NET*64 f32 = 12.8 MB (alias)
  char*  sm   = ws + 128000000ULL;
  float* degf = (float*)(sm);
  float* degr = (float*)(sm + 1000000);
  float* disf = (float*)(sm + 2000000);
  float* disr = (float*)(sm + 3000000);
  float* invf = (float*)(sm + 4000000);
  float* invr = (float*)(sm + 5000000);
  const int* row = eidx;
  const int* col = eidx + NEDGE;

  // encoders -> h0 lives in out[:, 0:64]
  enc1_inst<<<(N_INST * 32 + 255) / 256, 256, 0, stream>>>(x, enc1_W, enc1_b, tbuf);
  enc2_inst<<<(N_INST * 16 + 255) / 256, 256, 0, stream>>>(tbuf, enc2_W, enc2_b, out);
  enc1_net <<<(N_NET * 16 + 255) / 256, 256, 0, stream>>>(x_net, net1_W, net1_b, tnet);
  enc2_net <<<(N_NET * 16 + 255) / 256, 256, 0, stream>>>(tnet, net2_W, net2_b, out);

  // degree tables
  deg_init <<<(NTOT + 255) / 256, 256, 0, stream>>>(degf, degr);
  deg_accum<<<(NEDGE + 255) / 256, 256, 0, stream>>>(row, col, degf, degr);
  deg_fin  <<<(NTOT + 255) / 256, 256, 0, stream>>>(degf, degr, disf, disr, invf, invr);

  const int gemm_blocks = (NTOT / 16 + 3) / 4;
  const int edge_blocks = (NEDGE * 16 + 255) / 256;
  for (int l = 0; l < LAYERS; ++l) {
    const float* hbase = out + l * 64;  // read h_l from its JK slice (stride 256)
    conv_gemm<<<gemm_blocks, 128, 0, stream>>>(hbase, 256, conv_W + l * 4096,
                                               conv_b + l * 64, conv_rt + l * 64,
                                               invf, xbuf, agg, /*addmode=*/0);
    edge_scatter<<<edge_blocks, 256, 0, stream>>>(xbuf, agg, row, col, disf);
    conv_gemm<<<gemm_blocks, 128, 0, stream>>>(hbase, 256, re_W + l * 4096,
                                               re_b + l * 64, re_rt + l * 64,
                                               invr, xbuf, agg, /*addmode=*/1);
    edge_scatter<<<edge_blocks, 256, 0, stream>>>(xbuf, agg, col, row, disr);
    finalize<<<(NTOT * 32 + 255) / 256, 256, 0, stream>>>(agg, ln_g + l * 64,
                                                          ln_b + l * 64, out, l + 1);
  }
}